// ChebConv_81484119540505
// MI455X (gfx1250) — compile-verified
//
#include <hip/hip_runtime.h>
#include <math.h>

// ---------------------------------------------------------------------------
// ChebConv stack for MI455X (gfx1250, wave32).
//   dense path : V_WMMA_F32_16X16X4_F32, W panel staged in LDS (ds_load_b64),
//                8 waves/block sharing one 16-column panel.
//   sparse path: wave-per-edge gather + global_atomic_add_f32 scatter
//                (unsafeAtomicAdd), global_prefetch_b8 on next source row.
//   final layer (outd=1): wave-per-node dot product + shfl_xor reduction.
// ---------------------------------------------------------------------------

typedef __attribute__((ext_vector_type(2))) float v2f;
typedef __attribute__((ext_vector_type(8))) float v8f;

static const int NL = 11;
static const int DIMS_H[NL + 1] = {128, 64, 96, 128, 128, 128, 128, 128, 128, 96, 64, 1};

#define GEMM_WPB 8   // waves per GEMM block (256 threads)

// ------------------------------- utility kernels ---------------------------

__global__ __launch_bounds__(256) void k_fill0(float* __restrict__ p, size_t n) {
    size_t i = (size_t)blockIdx.x * blockDim.x + threadIdx.x;
    if (i < n) p[i] = 0.0f;
}

__global__ __launch_bounds__(256) void k_negcopy(float* __restrict__ o,
                                                 const float* __restrict__ a, size_t n) {
    size_t i = (size_t)blockIdx.x * blockDim.x + threadIdx.x;
    if (i < n) o[i] = -a[i];
}

__global__ __launch_bounds__(256) void k_degree(const long long* __restrict__ src,
                                                float* __restrict__ deg, int nE) {
    int e = blockIdx.x * blockDim.x + threadIdx.x;
    if (e < nE) unsafeAtomicAdd(&deg[(int)src[e]], 1.0f);
}

// deg -> dinv in place: dinv = deg>0 ? rsqrt(max(deg,1)) : 0
__global__ __launch_bounds__(256) void k_dinv(float* __restrict__ deg, int n) {
    int i = blockIdx.x * blockDim.x + threadIdx.x;
    if (i < n) {
        float d = deg[i];
        deg[i] = (d > 0.0f) ? rsqrtf(fmaxf(d, 1.0f)) : 0.0f;
    }
}

__global__ __launch_bounds__(256) void k_edgew(const long long* __restrict__ src,
                                               const long long* __restrict__ dst,
                                               const float* __restrict__ dinv,
                                               float* __restrict__ w, int nE) {
    int e = blockIdx.x * blockDim.x + threadIdx.x;
    if (e < nE) w[e] = -dinv[(int)src[e]] * dinv[(int)dst[e]];
}

// ------------------------- sparse propagation (SpMM) -----------------------
// tout[dst] += scale * w[e] * tin[src]; one wave per edge, lanes stride dim.
__global__ __launch_bounds__(256) void k_scatter(const long long* __restrict__ src,
                                                 const long long* __restrict__ dst,
                                                 const float* __restrict__ w,
                                                 const float* __restrict__ tin,
                                                 float* __restrict__ tout,
                                                 int nE, int dim, float scale) {
    const int lane   = threadIdx.x & 31;
    const int wid    = blockIdx.x * (blockDim.x >> 5) + (threadIdx.x >> 5);
    const int nwaves = gridDim.x * (blockDim.x >> 5);

    for (int e = wid; e < nE; e += nwaves) {
        const int   s  = (int)src[e];
        const int   d  = (int)dst[e];
        const float wt = scale * w[e];
        const float* __restrict__ pin  = tin  + (size_t)s * dim;
        float*       __restrict__ pout = tout + (size_t)d * dim;

        // prefetch the next edge's source row while we work on this one
        int en = e + nwaves;
        if (en < nE) {
            const float* pn = tin + (size_t)((int)src[en]) * dim;
            __builtin_prefetch(pn + lane * 4, 0, 1);   // -> global_prefetch_b8
        }

        for (int f = lane; f < dim; f += 32)
            unsafeAtomicAdd(&pout[f], wt * pin[f]);    // -> global_atomic_add_f32
    }
}

// ------------------------------ dense GEMM ---------------------------------
// C[rows,outd] (+)= A[rows,ind] @ W[ind,outd] via V_WMMA_F32_16X16X4_F32.
// Requires outd % 16 == 0, ind % 4 == 0, rows % 16 == 0.
// 8 waves/block, each wave one 16x16 tile; the block's 16-column W panel is
// staged in LDS pair-interleaved so a B fragment is one ds_load_b64.
// mode: 0 = init acc with bias, 1 = accumulate into C, 2 = accumulate + ReLU.
__global__ __launch_bounds__(256) void k_gemm(const float* __restrict__ A,
                                              const float* __restrict__ W,
                                              float* __restrict__ C,
                                              const float* __restrict__ bias,
                                              int rows, int ind, int outd, int mode) {
    __shared__ float lwb[128 * 16];   // [ind/2][16 cols][2 rows] , 8 KB max

    const int tid     = threadIdx.x;
    const int lane    = tid & 31;
    const int m16     = lane & 15;
    const int hi      = lane >> 4;    // half-wave select
    const int khalf   = hi << 1;      // K offset within 4-step: 0 or 2
    const int colbase = blockIdx.y << 4;

    // cooperative panel load: lwb[(j>>1)*32 + c*2 + (j&1)] = W[j][colbase+c]
    for (int t = tid; t < ind * 16; t += 256) {
        const int j = t >> 4;
        const int c = t & 15;
        lwb[(j >> 1) * 32 + c * 2 + (j & 1)] = W[(size_t)j * outd + colbase + c];
    }
    __syncthreads();

    const int rowtile = blockIdx.x * GEMM_WPB + (tid >> 5);
    const int rowbase = rowtile << 4;
    if (rowbase >= rows) return;      // wave-uniform: EXEC stays all-ones below

    v8f acc;
    if (mode == 0) {
        const float bv = bias[colbase + m16];
#pragma unroll
        for (int i = 0; i < 8; ++i) acc[i] = bv;
    } else {
#pragma unroll
        for (int i = 0; i < 8; ++i)
            acc[i] = C[(size_t)(rowbase + i + (hi << 3)) * outd + colbase + m16];
    }

    // A fragment (16x4 f32): lane m16 = row M; v0 = K khalf, v1 = K khalf+1
    const float* __restrict__ arow = A + (size_t)(rowbase + m16) * ind;
    const float* __restrict__ lb   = lwb + m16 * 2;

#pragma unroll 4
    for (int kk = 0; kk < ind; kk += 4) {
        const v2f a = *(const v2f*)(arow + kk + khalf);                 // global_load_b64
        const v2f b = *(const v2f*)(lb + ((kk + khalf) >> 1) * 32);     // ds_load_b64
        acc = __builtin_amdgcn_wmma_f32_16x16x4_f32(
                  /*neg_a=*/false, a, /*neg_b=*/false, b,
                  /*c_mod=*/(short)0, acc, /*reuse_a=*/false, /*reuse_b=*/false);
    }

#pragma unroll
    for (int i = 0; i < 8; ++i) {
        float v = acc[i];
        if (mode == 2) v = fmaxf(v, 0.0f);
        C[(size_t)(rowbase + i + (hi << 3)) * outd + colbase + m16] = v;
    }
}

// -------------------- final layer (outd == 1): mat-vec ---------------------
// out[n] (+)= dot(A[n,:], w[:]) ; wave per node, shfl_xor reduction.
__global__ __launch_bounds__(256) void k_dotcol(const float* __restrict__ A,
                                                const float* __restrict__ w,
                                                float* __restrict__ out,
                                                const float* __restrict__ bias,
                                                int rows, int ind, int mode) {
    const int lane = threadIdx.x & 31;
    const int node = blockIdx.x * (blockDim.x >> 5) + (threadIdx.x >> 5);
    if (node >= rows) return;

    const float* __restrict__ arow = A + (size_t)node * ind;
    float s = 0.0f;
    for (int f = lane; f < ind; f += 32) s += arow[f] * w[f];
#pragma unroll
    for (int off = 16; off; off >>= 1) s += __shfl_xor(s, off, 32);
    if (lane == 0)
        out[node] = (mode == 0) ? (s + bias[0]) : (out[node] + s);
}

// ------------------------------- launcher ----------------------------------

extern "C" void kernel_launch(void* const* d_in, const int* in_sizes, int n_in,
                              void* d_out, int out_size, void* d_ws, size_t ws_size,
                              hipStream_t stream) {
    (void)n_in; (void)out_size; (void)ws_size;

    const int N = in_sizes[0] / DIMS_H[0];
    const int E = in_sizes[1] / 2;

    const float*     x   = (const float*)d_in[0];
    const long long* ei  = (const long long*)d_in[1];   // edge_index int64 [2,E]
    const long long* src = ei;
    const long long* dst = ei + E;

    const float* Ws[NL];
    const float* Bs[NL];
    for (int i = 0; i < NL; ++i) {
        Ws[i] = (const float*)d_in[2 + i];        // [K, in, out]
        Bs[i] = (const float*)d_in[2 + NL + i];   // [out]
    }

    // workspace carve-out (floats): dinv[N] | w[E] | 4 slabs of N*128
    float* ws   = (float*)d_ws;
    float* dinv = ws;                  ws += N;
    float* ew   = ws;                  ws += E;
    const size_t slab = (size_t)N * 128;
    float* H = ws;                     ws += slab;   // T0 / incoming h
    float* P = ws;                     ws += slab;   // T1 (then T4)
    float* Q = ws;                     ws += slab;   // T2
    float* O = ws;                     ws += slab;   // layer output

    // ---- normalization: deg -> dinv -> edge weights ----
    k_fill0 <<<(N + 255) / 256, 256, 0, stream>>>(dinv, (size_t)N);
    k_degree<<<(E + 255) / 256, 256, 0, stream>>>(src, dinv, E);
    k_dinv  <<<(N + 255) / 256, 256, 0, stream>>>(dinv, N);
    k_edgew <<<(E + 255) / 256, 256, 0, stream>>>(src, dst, dinv, ew, E);

    // h = x (x is const; the T0 slab gets recycled mid-layer)
    hipMemcpyAsync(H, x, (size_t)N * DIMS_H[0] * sizeof(float),
                   hipMemcpyDeviceToDevice, stream);

    const int SC_BLOCKS  = 4096;                       // 8 waves each, grid-stride
    const int rowTiles   = N / 16;                     // 6250
    const int gemmBlocks = (rowTiles + GEMM_WPB - 1) / GEMM_WPB;
    const int dotBlocks  = (N + 7) / 8;                // 8 waves/block, wave/node

    for (int L = 0; L < NL; ++L) {
        const int ind  = DIMS_H[L];
        const int outd = DIMS_H[L + 1];
        const size_t wstride = (size_t)ind * outd;     // W[k] slab stride
        const bool last = (L == NL - 1);
        float* Cc = last ? (float*)d_out : O;

        const dim3 gg(gemmBlocks, outd / 16);
        const size_t nel = (size_t)N * ind;
        const int  gfill = (int)((nel + 255) / 256);

        // helper macro: emit GEMM (or final-layer dot) for slab Tk with W[k]
#define CHEB_MM(Tk, kidx, md)                                                        \
        do {                                                                         \
            if (last) k_dotcol<<<dotBlocks, 256, 0, stream>>>(                       \
                          (Tk), Ws[L] + (kidx) * wstride, Cc, Bs[L], N, ind,         \
                          (md) == 0 ? 0 : 1);                                        \
            else      k_gemm<<<gg, 256, 0, stream>>>(                                \
                          (Tk), Ws[L] + (kidx) * wstride, Cc, Bs[L], N, ind, outd,   \
                          (md));                                                     \
        } while (0)

        // k = 0 : out = T0 @ W0 + b
        CHEB_MM(H, 0, 0);

        // k = 1 : T1 = prop(T0)
        k_fill0  <<<gfill, 256, 0, stream>>>(P, nel);
        k_scatter<<<SC_BLOCKS, 256, 0, stream>>>(src, dst, ew, H, P, E, ind, 1.0f);
        CHEB_MM(P, 1, 1);

        // k = 2 : T2 = 2*prop(T1) - T0
        k_negcopy<<<gfill, 256, 0, stream>>>(Q, H, nel);
        k_scatter<<<SC_BLOCKS, 256, 0, stream>>>(src, dst, ew, P, Q, E, ind, 2.0f);
        CHEB_MM(Q, 2, 1);

        // k = 3 : T3 = 2*prop(T2) - T1   (T0 slab H is free -> holds T3)
        k_negcopy<<<gfill, 256, 0, stream>>>(H, P, nel);
        k_scatter<<<SC_BLOCKS, 256, 0, stream>>>(src, dst, ew, Q, H, E, ind, 2.0f);
        CHEB_MM(H, 3, 1);

        // k = 4 : T4 = 2*prop(T3) - T2   (T1 slab P is free -> holds T4)
        k_negcopy<<<gfill, 256, 0, stream>>>(P, Q, nel);
        k_scatter<<<SC_BLOCKS, 256, 0, stream>>>(src, dst, ew, H, P, E, ind, 2.0f);
        CHEB_MM(P, 4, last ? 1 : 2);   // ReLU on all but the final layer
#undef CHEB_MM

        // ping-pong: next layer's h is this layer's output
        if (L < NL - 1) { float* t = H; H = O; O = t; }
    }
}